// GNN_48653389529562
// MI455X (gfx1250) — compile-verified
//
#include <hip/hip_runtime.h>
#include <hip/hip_bf16.h>

typedef __attribute__((ext_vector_type(2))) float v2f;
typedef __attribute__((ext_vector_type(8))) float v8f;

// ---------------------------------------------------------------------------
// Degree / normalization kernels
// ---------------------------------------------------------------------------
__global__ void k_fill1(float* __restrict__ p, int n) {
    int i = blockIdx.x * blockDim.x + threadIdx.x;
    if (i < n) p[i] = 1.0f;                      // self-loop contributes 1 to deg
}

__global__ void k_deg_count(const int* __restrict__ dst, float* __restrict__ deg, int E) {
    int e = blockIdx.x * blockDim.x + threadIdx.x;
    if (e < E) atomicAdd(deg + dst[e], 1.0f);
}

__global__ void k_rsqrt_inplace(float* __restrict__ p, int n) {
    int i = blockIdx.x * blockDim.x + threadIdx.x;
    if (i < n) p[i] = rsqrtf(p[i]);
}

// ---------------------------------------------------------------------------
// FP32 WMMA GEMM: D[MxN] = A[MxK] @ B[KxN], row-major, M%16==0.
// K,N compile-time so the K-loop fully unrolls into straight-line WMMAs.
// One wave computes one 16x16 tile via V_WMMA_F32_16X16X4_F32 (K/4 of them).
// A-frag (16x4 f32): lane l -> row m=l&15; VGPR0 holds K=(l>=16?2:0), VGPR1 K+1.
// B-frag (4x16 f32): lane l -> col n=l&15; VGPR0 holds K=(l>=16?2:0), VGPR1 K+1.
// C/D (16x16 f32):   VGPR r, lane l -> row r + 8*(l>=16), col l&15.
// ---------------------------------------------------------------------------
template <int K, int N>
__global__ __launch_bounds__(256) void gemm_wmma_f32(
    const float* __restrict__ A, const float* __restrict__ B, float* __restrict__ D,
    int M)
{
    const int lane  = threadIdx.x & 31;
    const int wave  = threadIdx.x >> 5;
    constexpr int tilesN = N >> 4;
    const int tilesTotal = (M >> 4) * tilesN;
    const int tile = blockIdx.x * (256 >> 5) + wave;
    if (tile >= tilesTotal) return;              // wave-uniform branch: EXEC stays all-1s

    const int tm = tile / tilesN;
    const int tn = tile - tm * tilesN;
    const int rowBase = tm << 4;
    const int colBase = tn << 4;
    const int mn    = lane & 15;                 // row (A) / col (B,D)
    const int khalf = (lane >> 4) << 1;          // 0 or 2

    const float* ap0 = A + (size_t)(rowBase + mn) * K + khalf;
    const float* bp0 = B + (size_t)khalf * N + colBase + mn;

    v8f c = {};
#pragma unroll
    for (int k = 0; k < K; k += 4) {
        v2f a, b;
        a.x = ap0[k];
        a.y = ap0[k + 1];
        b.x = bp0[k * N];
        b.y = bp0[(k + 1) * N];
        c = __builtin_amdgcn_wmma_f32_16x16x4_f32(
                /*neg_a=*/false, a, /*neg_b=*/false, b,
                /*c_mod=*/(short)0, c, /*reuse_a=*/false, /*reuse_b=*/false);
    }

    const int rhi  = (lane >> 4) << 3;           // 0 or 8
    const int dcol = colBase + mn;
#pragma unroll
    for (int vr = 0; vr < 8; ++vr) {
        D[(size_t)(rowBase + rhi + vr) * N + dcol] = c[vr];
    }
}

// ---------------------------------------------------------------------------
// Aggregation kernels
// ---------------------------------------------------------------------------
template <int F>
__global__ void k_selfloop_init(const float* __restrict__ h, const float* __restrict__ dinv,
                                float* __restrict__ agg, int n) {
    long long idx = (long long)blockIdx.x * blockDim.x + threadIdx.x;
    if (idx < (long long)n * F) {
        int i = (int)(idx / F);
        float w = dinv[i];
        agg[idx] = h[idx] * w * w;
    }
}

__global__ void k_selfloop_bias_init16(const float* __restrict__ h, const float* __restrict__ dinv,
                                       const float* __restrict__ bias, float* __restrict__ out, int n) {
    long long idx = (long long)blockIdx.x * blockDim.x + threadIdx.x;
    if (idx < (long long)n * 16) {
        int i = (int)(idx >> 4);
        int f = (int)(idx & 15);
        float w = dinv[i];
        out[idx] = h[idx] * w * w + bias[f];
    }
}

__global__ void k_bias_relu64(const float* __restrict__ agg, const float* __restrict__ bias,
                              float* __restrict__ out, int n) {
    long long idx = (long long)blockIdx.x * blockDim.x + threadIdx.x;
    if (idx < (long long)n * 64) {
        int f = (int)(idx & 63);
        out[idx] = fmaxf(agg[idx] + bias[f], 0.0f);
    }
}

// One thread handles one edge x one 4-feature chunk: float4 gather + 4 atomics.
// Gathers and atomic scatters stay L2-resident (h/agg buffers << 192MB L2).
template <int CHUNKS>  // CHUNKS = F/4
__global__ void k_edge_agg(const int* __restrict__ src, const int* __restrict__ dst,
                           const float* __restrict__ dinv, const float* __restrict__ h,
                           float* __restrict__ agg, int E) {
    long long tid = (long long)blockIdx.x * blockDim.x + threadIdx.x;
    if (tid >= (long long)E * CHUNKS) return;
    int e = (int)(tid / CHUNKS);
    int q = (int)(tid - (long long)e * CHUNKS);
    int s = src[e], d = dst[e];
    float w = dinv[s] * dinv[d];
    const float4 hv = reinterpret_cast<const float4*>(h + (size_t)s * (CHUNKS * 4))[q];
    float* op = agg + (size_t)d * (CHUNKS * 4) + q * 4;
    atomicAdd(op + 0, hv.x * w);
    atomicAdd(op + 1, hv.y * w);
    atomicAdd(op + 2, hv.z * w);
    atomicAdd(op + 3, hv.w * w);
}

// ---------------------------------------------------------------------------
// Launch
// ---------------------------------------------------------------------------
extern "C" void kernel_launch(void* const* d_in, const int* in_sizes, int n_in,
                              void* d_out, int out_size, void* d_ws, size_t ws_size,
                              hipStream_t stream) {
    const float* x  = (const float*)d_in[0];   // [N, 64]
    const int*   ei = (const int*)d_in[1];     // [2, E]
    const float* W1 = (const float*)d_in[2];   // [64, 64]
    const float* b1 = (const float*)d_in[3];   // [64]
    const float* W2 = (const float*)d_in[4];   // [64, 16]
    const float* b2 = (const float*)d_in[5];   // [16]
    float*       out = (float*)d_out;          // [N, 16]

    const int N = in_sizes[0] / 64;
    const int E = in_sizes[1] / 2;
    const int* src = ei;
    const int* dst = ei + E;

    char* ws = (char*)d_ws;
    size_t off = 0;
    auto wsAlloc = [&](size_t bytes) -> float* {
        float* p = (float*)(ws + off);
        off += (bytes + 255) & ~(size_t)255;
        return p;
    };
    float* dinv = wsAlloc((size_t)N * sizeof(float));          // deg -> dinv in place
    float* h1   = wsAlloc((size_t)N * 64 * sizeof(float));     // x@W1, later relu(agg1+b1)
    float* agg1 = wsAlloc((size_t)N * 64 * sizeof(float));     // layer-1 aggregation
    float* h2   = wsAlloc((size_t)N * 16 * sizeof(float));     // hrelu@W2
    (void)ws_size; (void)n_in; (void)out_size;

    const int TPB = 256;
    auto blocks = [&](long long work) -> unsigned int {
        return (unsigned int)((work + TPB - 1) / TPB);
    };

    // --- normalization: deg (with self loops) -> dinv ---
    k_fill1<<<blocks(N), TPB, 0, stream>>>(dinv, N);
    k_deg_count<<<blocks(E), TPB, 0, stream>>>(dst, dinv, E);
    k_rsqrt_inplace<<<blocks(N), TPB, 0, stream>>>(dinv, N);

    // --- layer 1: h1 = x @ W1 (WMMA fp32) ---
    {
        int tiles = (N >> 4) * (64 >> 4);
        gemm_wmma_f32<64, 64><<<(tiles + 7) / 8, 256, 0, stream>>>(x, W1, h1, N);
    }
    k_selfloop_init<64><<<blocks((long long)N * 64), TPB, 0, stream>>>(h1, dinv, agg1, N);
    k_edge_agg<16><<<blocks((long long)E * 16), TPB, 0, stream>>>(src, dst, dinv, h1, agg1, E);
    k_bias_relu64<<<blocks((long long)N * 64), TPB, 0, stream>>>(agg1, b1, h1, N);

    // --- layer 2: h2 = relu(h1) @ W2 (WMMA fp32) ---
    {
        int tiles = (N >> 4) * (16 >> 4);
        gemm_wmma_f32<64, 16><<<(tiles + 7) / 8, 256, 0, stream>>>(h1, W2, h2, N);
    }
    k_selfloop_bias_init16<<<blocks((long long)N * 16), TPB, 0, stream>>>(h2, dinv, b2, out, N);
    k_edge_agg<4><<<blocks((long long)E * 4), TPB, 0, stream>>>(src, dst, dinv, h2, out, E);
}